// ChebGCN_13975823581724
// MI455X (gfx1250) — compile-verified
//
#include <hip/hip_runtime.h>
#include <hip/hip_bf16.h>

#define NN 100000          // N_NODES
#define NE 3200000         // N_EDGES
#define FH 128             // IN_F == H
#define NC 8               // N_CLASSES
#define NL 5               // N_LAYERS

typedef float v2f __attribute__((ext_vector_type(2)));
typedef float v8f __attribute__((ext_vector_type(8)));

#define GEMM_INIT_BIAS 1
#define GEMM_ACT 2

// ---------------- degree / norm ----------------
__global__ void cheb_deg_kernel(const int* __restrict__ dst, float* __restrict__ deg) {
    int e = blockIdx.x * blockDim.x + threadIdx.x;
    if (e < NE) atomicAdd(&deg[dst[e]], 1.0f);
}

__global__ void cheb_norm_kernel(float* __restrict__ norm) {
    int i = blockIdx.x * blockDim.x + threadIdx.x;
    if (i < NN) {
        float d = norm[i];
        norm[i] = rsqrtf(fmaxf(d, 1.0f));
    }
}

// ---------------- edge gather -> atomic scatter ----------------
// one wave per edge, 32 lanes x float4 = 128 features
__global__ void cheb_scatter_kernel(const float* __restrict__ v,
                                    const float* __restrict__ norm,
                                    const int* __restrict__ src,
                                    const int* __restrict__ dst,
                                    float* __restrict__ agg) {
    int gid  = blockIdx.x * blockDim.x + threadIdx.x;
    int e    = gid >> 5;
    int lane = threadIdx.x & 31;
    if (e >= NE) return;
    int s = src[e];
    int d = dst[e];
    float ns = norm[s];
    float4 val = ((const float4*)(v + (size_t)s * FH))[lane];
    float* ap = agg + (size_t)d * FH + lane * 4;
    atomicAdd(ap + 0, ns * val.x);
    atomicAdd(ap + 1, ns * val.y);
    atomicAdd(ap + 2, ns * val.z);
    atomicAdd(ap + 3, ns * val.w);
}

// ---------------- Chebyshev combine: out = alpha*norm[r]*agg - sub ----------------
__global__ void cheb_combine_kernel(const float* __restrict__ agg,
                                    const float* __restrict__ norm,
                                    const float* __restrict__ sub,
                                    float* __restrict__ out,
                                    float alpha) {
    int idx = blockIdx.x * blockDim.x + threadIdx.x;   // NN*32 float4 items
    if (idx >= NN * 32) return;
    int r = idx >> 5;
    float a = alpha * norm[r];
    float4 g = ((const float4*)agg)[idx];
    float4 o;
    if (sub) {
        float4 sv = ((const float4*)sub)[idx];
        o.x = fmaf(a, g.x, -sv.x);
        o.y = fmaf(a, g.y, -sv.y);
        o.z = fmaf(a, g.z, -sv.z);
        o.w = fmaf(a, g.w, -sv.w);
    } else {
        o.x = a * g.x; o.y = a * g.y; o.z = a * g.z; o.w = a * g.w;
    }
    ((float4*)out)[idx] = o;
}

// ---------------- fp32 WMMA GEMM: Out(N x 128) (+)= A(N x 128) @ Wb(128 x 128) ----------------
// block = 256 threads = 8 waves; block owns rows [16*bx, 16*bx+16), wave w owns cols [16w,16w+16)
__global__ __launch_bounds__(256)
void cheb_gemm_wmma_kernel(const float* __restrict__ A,
                           const float* __restrict__ Wb,
                           const float* __restrict__ bias,
                           float* __restrict__ Out,
                           int flags) {
    __shared__ float As[16 * 132];   // stride 132 -> conflict-free ds_load_b64 fragments

    const int tid  = threadIdx.x;
    const int lane = tid & 31;
    const int half = lane >> 4;      // 0: K 0,1 / M 0..7   1: K 2,3 / M 8..15
    const int lidx = lane & 15;
    const int wave = tid >> 5;
    const int m0   = blockIdx.x * 16;
    const int n0   = wave * 16;

    // cooperative stage of the 16x128 A slab
    {
        int r = tid >> 4;            // 0..15
        int c = (tid & 15) * 8;      // 0..120
        const float4* g = (const float4*)(A + (size_t)(m0 + r) * FH + c);
        float4 v0 = g[0], v1 = g[1];
        float* s = As + r * 132 + c;
        s[0] = v0.x; s[1] = v0.y; s[2] = v0.z; s[3] = v0.w;
        s[4] = v1.x; s[5] = v1.y; s[6] = v1.z; s[7] = v1.w;
    }
    __syncthreads();

    v8f c;
    if (flags & GEMM_INIT_BIAS) {
        float bv = bias[n0 + lidx];
#pragma unroll
        for (int i = 0; i < 8; ++i) c[i] = bv;
    } else {
        const float* op = Out + (size_t)(m0 + 8 * half) * FH + n0 + lidx;
#pragma unroll
        for (int i = 0; i < 8; ++i) c[i] = op[(size_t)i * FH];
    }

    // A frag: lane holds row lidx, K = kb + 2*half + {0,1}
    // B frag: lane holds col n0+lidx, same K pair
    const float* ap = As + lidx * 132 + 2 * half;
    const float* wp = Wb + (size_t)(2 * half) * FH + n0 + lidx;
#pragma unroll
    for (int kb = 0; kb < FH; kb += 4) {
        v2f a, b;
        a.x = ap[kb];
        a.y = ap[kb + 1];
        b.x = wp[(size_t)kb * FH];
        b.y = wp[(size_t)(kb + 1) * FH];
        c = __builtin_amdgcn_wmma_f32_16x16x4_f32(false, a, false, b, (short)0, c,
                                                  false, false);
    }

    float* op = Out + (size_t)(m0 + 8 * half) * FH + n0 + lidx;
    if (flags & GEMM_ACT) {
#pragma unroll
        for (int i = 0; i < 8; ++i) {
            float v = c[i];
            op[(size_t)i * FH] = v > 0.0f ? v : 0.01f * v;
        }
    } else {
#pragma unroll
        for (int i = 0; i < 8; ++i) op[(size_t)i * FH] = c[i];
    }
}

// ---------------- head: out(N x 8) = h(N x 128) @ Wout(128 x 8) + bout ----------------
__global__ void cheb_head_kernel(const float* __restrict__ h,
                                 const float* __restrict__ Wout,
                                 const float* __restrict__ bout,
                                 float* __restrict__ out) {
    int idx = blockIdx.x * blockDim.x + threadIdx.x;
    if (idx >= NN * NC) return;
    int node = idx >> 3;
    int cls  = idx & 7;
    const float* hp = h + (size_t)node * FH;
    float acc = bout[cls];
#pragma unroll 8
    for (int j = 0; j < FH; ++j) acc = fmaf(hp[j], Wout[j * NC + cls], acc);
    out[idx] = acc;
}

// ---------------- orchestration ----------------
extern "C" void kernel_launch(void* const* d_in, const int* in_sizes, int n_in,
                              void* d_out, int out_size, void* d_ws, size_t ws_size,
                              hipStream_t stream) {
    (void)in_sizes; (void)n_in; (void)out_size; (void)ws_size;
    const float* x    = (const float*)d_in[0];
    const int*   src  = (const int*)d_in[1];
    const int*   dst  = (const int*)d_in[2];
    const float* Wc   = (const float*)d_in[3];   // (5, 640, 128)
    const float* bc   = (const float*)d_in[4];   // (5, 128)
    const float* Wout = (const float*)d_in[5];   // (128, 8)
    const float* bout = (const float*)d_in[6];   // (8)
    float* out = (float*)d_out;

    const size_t NH = (size_t)NN * FH;
    float* ws   = (float*)d_ws;
    float* norm = ws;                 // NN floats (deg then norm, in place)
    float* agg  = ws + 100096;        // padded, 16B aligned
    float* tA   = agg + NH;
    float* tB   = tA + NH;
    float* hA   = tB + NH;
    float* hB   = hA + NH;

    const int scatterBlocks = (NE * 32) / 256;         // exact: 400000
    const int combBlocks    = (NN * 32 + 255) / 256;
    const int gemmBlocks    = NN / 16;                 // exact: 6250

    // degree -> norm = clip(deg,1)^-0.5
    hipMemsetAsync(norm, 0, NN * sizeof(float), stream);
    cheb_deg_kernel<<<(NE + 255) / 256, 256, 0, stream>>>(dst, norm);
    cheb_norm_kernel<<<(NN + 255) / 256, 256, 0, stream>>>(norm);

    const float* h_in = x;
    float* h_out = hA;
    for (int l = 0; l < NL; ++l) {
        const float* Wl = Wc + (size_t)l * (5 * FH) * FH;
        const float* bl = bc + (size_t)l * FH;
        const size_t WBLK = (size_t)FH * FH;   // 128x128 per Chebyshev order

        // t1 = lhat(h_in) = -norm*agg(h_in)   -> tA
        hipMemsetAsync(agg, 0, NH * sizeof(float), stream);
        cheb_scatter_kernel<<<scatterBlocks, 256, 0, stream>>>(h_in, norm, src, dst, agg);
        cheb_combine_kernel<<<combBlocks, 256, 0, stream>>>(agg, norm, nullptr, tA, -1.0f);

        // Out = bias + t0 @ W0 ;  Out += t1 @ W1
        cheb_gemm_wmma_kernel<<<gemmBlocks, 256, 0, stream>>>(h_in, Wl,        bl, h_out, GEMM_INIT_BIAS);
        cheb_gemm_wmma_kernel<<<gemmBlocks, 256, 0, stream>>>(tA,   Wl + WBLK, bl, h_out, 0);

        // t2 = -2*norm*agg(t1) - t0 -> tB
        hipMemsetAsync(agg, 0, NH * sizeof(float), stream);
        cheb_scatter_kernel<<<scatterBlocks, 256, 0, stream>>>(tA, norm, src, dst, agg);
        cheb_combine_kernel<<<combBlocks, 256, 0, stream>>>(agg, norm, h_in, tB, -2.0f);
        cheb_gemm_wmma_kernel<<<gemmBlocks, 256, 0, stream>>>(tB, Wl + 2 * WBLK, bl, h_out, 0);

        // t3 = -2*norm*agg(t2) - t1 -> tA (in-place over t1, elementwise-safe)
        hipMemsetAsync(agg, 0, NH * sizeof(float), stream);
        cheb_scatter_kernel<<<scatterBlocks, 256, 0, stream>>>(tB, norm, src, dst, agg);
        cheb_combine_kernel<<<combBlocks, 256, 0, stream>>>(agg, norm, tA, tA, -2.0f);
        cheb_gemm_wmma_kernel<<<gemmBlocks, 256, 0, stream>>>(tA, Wl + 3 * WBLK, bl, h_out, 0);

        // t4 = -2*norm*agg(t3) - t2 -> tB ; last pass fuses leaky_relu
        hipMemsetAsync(agg, 0, NH * sizeof(float), stream);
        cheb_scatter_kernel<<<scatterBlocks, 256, 0, stream>>>(tA, norm, src, dst, agg);
        cheb_combine_kernel<<<combBlocks, 256, 0, stream>>>(agg, norm, tB, tB, -2.0f);
        cheb_gemm_wmma_kernel<<<gemmBlocks, 256, 0, stream>>>(tB, Wl + 4 * WBLK, bl, h_out, GEMM_ACT);

        h_in  = h_out;
        h_out = (h_out == hA) ? hB : hA;
    }

    cheb_head_kernel<<<(NN * NC + 255) / 256, 256, 0, stream>>>(h_in, Wout, bout, out);
}